// CTCLayer_40621800685628
// MI455X (gfx1250) — compile-verified
//
#include <hip/hip_runtime.h>
#include <math.h>
#include <stdint.h>

#define NEGV (-1e30f)

// logaddexp(a,b) robust for NEGV sentinels: max + log1p(exp(-|a-b|))
__device__ __forceinline__ float lse2(float a, float b) {
  float m = fmaxf(a, b);
  return m + log1pf(expf(-fabsf(a - b)));
}

// CDNA5 async global->LDS copy (GV mode, 4 bytes/lane), tracked by ASYNCcnt.
__device__ __forceinline__ void async_load_b32(const float* gaddr, void* ldsaddr) {
  uint32_t lds = (uint32_t)(uintptr_t)ldsaddr;  // low 32 bits of flat LDS addr = LDS byte offset
  asm volatile("global_load_async_to_lds_b32 %0, %1, off"
               :: "v"(lds), "v"(gaddr)
               : "memory");
}

__device__ __forceinline__ void wait_async_le0() {
  asm volatile("s_wait_asynccnt 0" ::: "memory");
}
__device__ __forceinline__ void wait_async_le1() {
  asm volatile("s_wait_asynccnt 1" ::: "memory");
}

// One workgroup per batch element. blockDim.x == 288 (9 wave32s).
// Requires C == 256; states s = tid for tid < S = 2L+1 <= 257.
__global__ __launch_bounds__(288)
void ctc_loss_kernel(const int*   __restrict__ y_true,     // [B,L]
                     const float* __restrict__ y_pred,     // [B,T,C]
                     const int*   __restrict__ input_len,  // [B,1]
                     const int*   __restrict__ label_len,  // [B,1]
                     float*       __restrict__ out,        // [B]
                     int T, int C, int L)
{
  const int b     = blockIdx.x;
  const int tid   = threadIdx.x;
  const int S     = 2 * L + 1;
  const int blank = C - 1;
  const int lane  = tid & 31;
  const int wid   = tid >> 5;
  const bool act  = (tid < S);      // this thread owns CTC state s = tid
  const bool ldr  = (tid < 256);    // this thread loads/reduces class c = tid

  __shared__ float rowbuf[3][256];   // async-prefetched probability rows (depth-2 pipeline)
  __shared__ float vbuf[2][256];     // log(p + eps) per class, parity-buffered
  __shared__ float alpha[2][272];    // ping-pong forward variables
  __shared__ int   exts[272];        // extended label sequence (setup only)
  __shared__ float partials[2][8];   // per-wave row sums, parity-buffered

  // ---- extended labels: blank at even s, y_true[s>>1] at odd s ----
  if (act) exts[tid] = (tid & 1) ? y_true[(size_t)b * L + (tid >> 1)] : blank;
  __syncthreads();

  int myext = blank, myskip = 0;
  if (act) {
    myext = exts[tid];
    bool same_prev = (tid < 2) ? true : (myext == exts[tid - 2]);
    myskip = (myext != blank) && !same_prev;
  }

  const int    inlen = input_len[b];
  const float* yp    = y_pred + (size_t)b * T * C;
  const float  epsC  = (float)C * 1e-7f;

  // ---- prologue: prefetch rows 0 and 1 (each lane its own class element) ----
  if (ldr) {
    async_load_b32(yp + tid, &rowbuf[0][tid]);
    if (T > 1) async_load_b32(yp + (size_t)C + tid, &rowbuf[1][tid]);
  }

  float aown = NEGV;                 // this thread's alpha[s], kept in a register
  for (int t = 0; t < T; ++t) {
    const int rb = t % 3;
    const int pb = t & 1;

    // ---- phase 1: retire row t, issue row t+2, compute v and row sum ----
    if (t + 1 < T) wait_async_le1();   // oldest outstanding (row t) complete
    else           wait_async_le0();

    float ps = 0.0f;
    if (ldr) {
      if (t + 2 < T)
        async_load_b32(yp + (size_t)(t + 2) * C + tid, &rowbuf[(t + 2) % 3][tid]);
      float p = rowbuf[rb][tid];
      vbuf[pb][tid] = logf(p + 1e-7f);
      ps = p;
    }
    #pragma unroll
    for (int o = 16; o > 0; o >>= 1) ps += __shfl_xor(ps, o, 32);
    if (lane == 0 && wid < 8) partials[pb][wid] = ps;

    __syncthreads();                   // single barrier per time step

    // ---- phase 3: logZ + alpha stencil ----
    float tot = 0.0f;
    #pragma unroll
    for (int w = 0; w < 8; ++w) tot += partials[pb][w];
    const float logZ = logf(tot + epsC);

    const float* vb = vbuf[pb];
    const float* ap = alpha[pb];         // alpha after step t-1 (valid for t>=1)
    float*       an = alpha[pb ^ 1];

    if (act) {
      if (t == 0) {
        float e = vb[myext] - logZ;
        aown = (tid <= 1) ? e : NEGV;    // alpha0 set unconditionally (as in reference)
      } else if (t < inlen) {
        float a1 = (tid >= 1) ? ap[tid - 1] : NEGV;
        float a2 = (tid >= 2 && myskip) ? ap[tid - 2] : NEGV;
        float e  = vb[myext] - logZ;
        aown = lse2(lse2(aown, a1), a2) + e;
      }
      an[tid] = aown;
    }
  }

  __syncthreads();
  if (tid == 0) {
    int lab = label_len[b];
    lab = lab < 1 ? 1 : (lab > L ? L : lab);
    const float* af = alpha[T & 1];
    out[b] = -lse2(af[2 * lab - 1], af[2 * lab]);
  }
}

extern "C" void kernel_launch(void* const* d_in, const int* in_sizes, int n_in,
                              void* d_out, int out_size, void* d_ws, size_t ws_size,
                              hipStream_t stream) {
  const int*   y_true    = (const int*)d_in[0];
  const float* y_pred    = (const float*)d_in[1];
  const int*   input_len = (const int*)d_in[2];
  const int*   label_len = (const int*)d_in[3];
  float*       out       = (float*)d_out;

  const int B = in_sizes[2];                 // input_length has B elements
  const int L = in_sizes[0] / B;             // y_true is [B,L]
  const int C = 256;                         // kernel requires C == 256
  const int T = in_sizes[1] / (B * C);       // y_pred is [B,T,C]

  ctc_loss_kernel<<<dim3(B), dim3(288), 0, stream>>>(
      y_true, y_pred, input_len, label_len, out, T, C, L);
}